// Distance_25383256719961
// MI455X (gfx1250) — compile-verified
//
#include <hip/hip_runtime.h>

typedef __attribute__((ext_vector_type(2))) float v2f;
typedef __attribute__((ext_vector_type(4))) float v4f;
typedef __attribute__((ext_vector_type(8))) float v8f;

#define N_ROWS   8192
#define LDA      64        // input row stride in floats (8192 x 64)
#define TILES_1D (N_ROWS / 16)   // 512
#define WAVES_PER_BLOCK 8

// One wave32 computes one 16x16 tile of exp(-sqrt(clip(na_i + na_j - 2*A.A^T))).
__global__ __launch_bounds__(256) void
distance_exp_kernel(const float* __restrict__ in, float* __restrict__ out)
{
    const int lane = threadIdx.x & 31;
    const int wid  = blockIdx.x * WAVES_PER_BLOCK + (threadIdx.x >> 5);
    const int I    = wid >> 9;          // tile row   (0..511)
    const int J    = wid & (TILES_1D - 1); // tile col (0..511)

    const int half = lane >> 4;         // 0: lanes 0-15, 1: lanes 16-31
    const int l16  = lane & 15;

    const int rowA = I * 16 + l16;      // this lane's A-side row
    const int rowB = J * 16 + l16;      // this lane's B-side row (== output column index row)
    const float* __restrict__ pA = in + rowA * LDA;
    const float* __restrict__ pB = in + rowB * LDA;

    // ---- row norms (sum of squares over the 16-dim slice) ----
    float na_own = 0.0f;   // ||A[rowA]||^2, held by lane l16 (duplicated in both halves)
    float nb_own = 0.0f;   // ||A[rowB]||^2, this lane's output-column norm
#pragma unroll
    for (int k = 0; k < 16; k += 4) {
        v4f va = *(const v4f*)(pA + k);
        v4f vb = *(const v4f*)(pB + k);
        na_own += va.x * va.x + va.y * va.y + va.z * va.z + va.w * va.w;
        nb_own += vb.x * vb.x + vb.y * vb.y + vb.z * vb.z + vb.w * vb.w;
    }

    // ---- A . A^T for this tile via 4x V_WMMA_F32_16X16X4_F32 (K = 16) ----
    // A(16x4) layout: lanes 0-15 hold K=k0+0..1, lanes 16-31 hold K=k0+2..3 (one row per lane).
    // B(4x16) mirrors it with columns per lane, so the same float2 load serves both operands.
    v8f c = {};
#pragma unroll
    for (int k0 = 0; k0 < 16; k0 += 4) {
        const int kk = k0 + 2 * half;
        v2f a = *(const v2f*)(pA + kk);
        v2f b = *(const v2f*)(pB + kk);
        c = __builtin_amdgcn_wmma_f32_16x16x4_f32(
                /*neg_a=*/false, a, /*neg_b=*/false, b,
                /*c_mod=*/(short)0, c, /*reuse_a=*/false, /*reuse_b=*/false);
    }

    // ---- epilogue: Dsq = na_i + na_j - 2*dot; clip; sqrt; exp(-D); NT store ----
    // C/D layout: VGPR r, lanes 0-15 -> M=r, N=lane; lanes 16-31 -> M=r+8, N=lane-16.
    const int colOut = J * 16 + l16;
    const float log2e = 1.44269504088896340736f;
#pragma unroll
    for (int r = 0; r < 8; ++r) {
        const int m    = r + 8 * half;           // row within tile for this accumulator
        const float ni = __shfl(na_own, m, 32);  // norm of row I*16+m (held by lane m)
        float dsq = ni + nb_own - 2.0f * c[r];
        dsq = fminf(fmaxf(dsq, 1e-12f), 1e12f);
        const float d = __builtin_amdgcn_sqrtf(dsq);
        const float v = __builtin_amdgcn_exp2f(-d * log2e);
        __builtin_nontemporal_store(v, out + (size_t)(I * 16 + m) * N_ROWS + colOut);
    }
}

extern "C" void kernel_launch(void* const* d_in, const int* in_sizes, int n_in,
                              void* d_out, int out_size, void* d_ws, size_t ws_size,
                              hipStream_t stream)
{
    const float* in  = (const float*)d_in[0];
    float*       out = (float*)d_out;

    const int total_tiles = TILES_1D * TILES_1D;              // 262144
    const int blocks      = total_tiles / WAVES_PER_BLOCK;    // 32768
    distance_exp_kernel<<<blocks, 32 * WAVES_PER_BLOCK, 0, stream>>>(in, out);
}